// RNet_30279519437509
// MI455X (gfx1250) — compile-verified
//
#include <hip/hip_runtime.h>
#include <hip/hip_bf16.h>
#include <math.h>

typedef __attribute__((ext_vector_type(2))) float v2f;
typedef __attribute__((ext_vector_type(8))) float v8f;

#define OSC 2048
#define MHID 4096

// ---- workspace layout (float offsets, 16B aligned) ----
#define WS_XM    0        // 4101 -> 4104
#define WS_XK1   4104     // 4097 -> 4104
#define WS_XK2   8208
#define WS_XF1   12312    // 2049 -> 2056
#define WS_XF2   14368
#define WS_XH1   16424    // 4097 -> 4104
#define WS_XH2   20528
#define WS_MNEW  24632    // 4096
#define WS_OUTM  28728    // 4096
#define WS_H1N   32824    // 2048
#define WS_H2N   34872
#define WS_K1N   36920
#define WS_K2N   38968
#define WS_F1N   41016
#define WS_F2N   43064

__device__ __forceinline__ float sigmf(float x) {
    return 1.0f / (1.0f + __expf(-x));
}

// fp32 WMMA used as a 16-row dot-product engine.
// K-permutation trick: per K-step of 8, lane l loads 4 contiguous floats at
// offset (8t + 4*(lane>>4)); chunk0 uses elements {0,1}, chunk1 uses {2,3}.
// A rows are (lane&15); B holds the vector -- only lanes 0 and 16 feed
// column N=0 of B, which is the only D column read. The C accumulators carry
// the running sums so the K-reduction happens in the matrix pipe.
// Main trip covers 64 K-values: ALL 16 b128 loads are issued into distinct
// registers first (deep MLP against ~HBM latency), then 16 WMMAs drain them
// across two independent accumulator chains. Weights stream non-temporally
// (one-shot, 873MB total >> 192MB L2); vectors stay regular-temporal.
__device__ __forceinline__ v8f wmma_dot_rows(const float* __restrict__ W,
                                             const float* __restrict__ vec,
                                             size_t row0, int K, int kbeg, int kend8,
                                             int lane) {
    const int m  = lane & 15;
    const int hh = lane >> 4;
    const float* wp = W + (row0 + (size_t)m) * (size_t)K + 4 * hh;
    const float* xp = vec + 4 * hh;
    v8f c0 = {}, c1 = {};
    int k = kbeg;
    for (; k + 64 <= kend8; k += 64) {
        v2f WA[16];
        v2f XB[16];
#pragma unroll
        for (int u = 0; u < 8; ++u) {
            const int kk = k + 8 * u;
            float a0 = __builtin_nontemporal_load(wp + kk + 0);
            float a1 = __builtin_nontemporal_load(wp + kk + 1);
            float a2 = __builtin_nontemporal_load(wp + kk + 2);
            float a3 = __builtin_nontemporal_load(wp + kk + 3);
            float b0 = xp[kk + 0], b1 = xp[kk + 1], b2 = xp[kk + 2], b3 = xp[kk + 3];
            WA[2 * u + 0] = (v2f){a0, a1};
            WA[2 * u + 1] = (v2f){a2, a3};
            XB[2 * u + 0] = (v2f){b0, b1};
            XB[2 * u + 1] = (v2f){b2, b3};
        }
#pragma unroll
        for (int u = 0; u < 8; ++u) {
            c0 = __builtin_amdgcn_wmma_f32_16x16x4_f32(false, WA[2 * u + 0], false, XB[2 * u + 0],
                                                       (short)0, c0, false, false);
            c1 = __builtin_amdgcn_wmma_f32_16x16x4_f32(false, WA[2 * u + 1], false, XB[2 * u + 1],
                                                       (short)0, c1, false, false);
        }
    }
    // remainder in 8-K steps
    for (; k < kend8; k += 8) {
        float a0 = __builtin_nontemporal_load(wp + k + 0);
        float a1 = __builtin_nontemporal_load(wp + k + 1);
        float a2 = __builtin_nontemporal_load(wp + k + 2);
        float a3 = __builtin_nontemporal_load(wp + k + 3);
        float b0 = xp[k + 0], b1 = xp[k + 1], b2 = xp[k + 2], b3 = xp[k + 3];
        v2f A0 = {a0, a1}, B0 = {b0, b1};
        v2f A1 = {a2, a3}, B1 = {b2, b3};
        c0 = __builtin_amdgcn_wmma_f32_16x16x4_f32(false, A0, false, B0, (short)0, c0, false, false);
        c1 = __builtin_amdgcn_wmma_f32_16x16x4_f32(false, A1, false, B1, (short)0, c1, false, false);
    }
    return c0 + c1;
}

// Fused GRU step for a 16-unit tile. 6 waves = {ih,hh} x {r,z,n} GEMV row-sets.
__global__ void __launch_bounds__(192)
gru16_kernel(const float* __restrict__ w_ih, const float* __restrict__ w_hh,
             const float* __restrict__ b_ih, const float* __restrict__ b_hh,
             const float* __restrict__ xin, const float* __restrict__ h,
             float* __restrict__ h_new, int H, int Kin) {
    __shared__ float gsum[6][16];
    __shared__ float gtail[6][16];

    const int tid  = threadIdx.x;
    const int wave = tid >> 5;          // 0..5
    const int lane = tid & 31;
    const int src  = wave / 3;          // 0 = ih, 1 = hh
    const int gate = wave % 3;          // 0 = r, 1 = z, 2 = n

    const float* W   = src ? w_hh : w_ih;
    const float* vec = src ? h    : xin;
    const int    K   = src ? H    : Kin;
    const size_t row0 = (size_t)gate * H + (size_t)blockIdx.x * 16;

    const int K8 = K & ~7;
    v8f c = wmma_dot_rows(W, vec, row0, K, 0, K8, lane);

    if ((lane & 15) == 0) {
        const int half = lane >> 4;
#pragma unroll
        for (int i = 0; i < 8; ++i) gsum[wave][half * 8 + i] = c[i];
    }
    if (lane < 16) {
        const float* wr = W + (row0 + (size_t)lane) * (size_t)K;
        float t = 0.0f;
        for (int k = K8; k < K; ++k) t += wr[k] * vec[k];
        gtail[wave][lane] = t;
    }
    __syncthreads();

    if (tid < 16) {
        const int u = tid;
        const int j = blockIdx.x * 16 + u;
        float ir = gsum[0][u] + gtail[0][u] + b_ih[j];
        float iz = gsum[1][u] + gtail[1][u] + b_ih[H + j];
        float in_ = gsum[2][u] + gtail[2][u] + b_ih[2 * H + j];
        float hr = gsum[3][u] + gtail[3][u] + b_hh[j];
        float hz = gsum[4][u] + gtail[4][u] + b_hh[H + j];
        float hn = gsum[5][u] + gtail[5][u] + b_hh[2 * H + j];
        float r = sigmf(ir + hr);
        float z = sigmf(iz + hz);
        float n = tanhf(in_ + r * hn);
        h_new[j] = (1.0f - z) * n + z * h[j];
    }
}

// 16 output rows per block, 8 waves split K; WMMA accumulators per wave.
__global__ void __launch_bounds__(256)
gemv16_kernel(const float* __restrict__ W, const float* __restrict__ bias,
              const float* __restrict__ x, float* __restrict__ out, int K) {
    __shared__ float gsum[8][16];
    __shared__ float gtail[8][16];

    const int tid  = threadIdx.x;
    const int wave = tid >> 5;          // 0..7
    const int lane = tid & 31;
    const size_t row0 = (size_t)blockIdx.x * 16;

    const int chunk = ((K + 63) / 64) * 8;       // per-wave K slice, multiple of 8
    int start = wave * chunk; if (start > K) start = K;
    int end   = start + chunk; if (end > K) end = K;
    const int end8 = start + ((end - start) & ~7);

    v8f c = wmma_dot_rows(W, x, row0, K, start, end8, lane);

    if ((lane & 15) == 0) {
        const int half = lane >> 4;
#pragma unroll
        for (int i = 0; i < 8; ++i) gsum[wave][half * 8 + i] = c[i];
    }
    if (lane < 16) {
        const float* wr = W + (row0 + (size_t)lane) * (size_t)K;
        float t = 0.0f;
        for (int k = end8; k < end; ++k) t += wr[k] * x[k];
        gtail[wave][lane] = t;
    }
    __syncthreads();

    if (tid < 16) {
        float s = 0.0f;
#pragma unroll
        for (int w = 0; w < 8; ++w) s += gsum[w][tid] + gtail[w][tid];
        out[row0 + tid] = s + bias[row0 + tid];
    }
}

// Build xm, xk1, xk2, xf1, xf2 concatenated input vectors.
__global__ void prep1_kernel(const float* __restrict__ x,
                             const float* __restrict__ h1s, const float* __restrict__ h2s,
                             const float* __restrict__ f1s, const float* __restrict__ f2s,
                             float* __restrict__ ws) {
    float* xm  = ws + WS_XM;
    float* xk1 = ws + WS_XK1;
    float* xk2 = ws + WS_XK2;
    float* xf1 = ws + WS_XF1;
    float* xf2 = ws + WS_XF2;
    const int i = blockIdx.x * blockDim.x + threadIdx.x;
    const int stride = gridDim.x * blockDim.x;
    if (i == 0) {
        xm[0] = x[0]; xm[1] = x[1]; xm[2] = x[8]; xm[3] = x[9]; xm[4] = x[10];
        xk1[0] = x[3]; xk2[0] = x[6];
        xf1[0] = x[4]; xf2[0] = x[7];
    }
    for (int j = i; j < OSC; j += stride) {
        float a = h1s[j], b = h2s[j], c = f1s[j], d = f2s[j];
        xm[5 + j]        = a;
        xm[5 + OSC + j]  = b;
        xk1[1 + j]       = a;
        xk1[1 + OSC + j] = c;
        xk2[1 + j]       = b;
        xk2[1 + OSC + j] = d;
        xf1[1 + j]       = c;
        xf2[1 + j]       = d;
    }
}

// Build xh1, xh2 (need out_m).
__global__ void prep2_kernel(const float* __restrict__ x,
                             const float* __restrict__ k1s, const float* __restrict__ k2s,
                             float* __restrict__ ws) {
    float* xh1 = ws + WS_XH1;
    float* xh2 = ws + WS_XH2;
    const float* outm = ws + WS_OUTM;
    const int i = blockIdx.x * blockDim.x + threadIdx.x;
    const int stride = gridDim.x * blockDim.x;
    if (i == 0) { xh1[0] = x[2]; xh2[0] = x[5]; }
    for (int j = i; j < OSC; j += stride) {
        xh1[1 + j]       = outm[j];
        xh1[1 + OSC + j] = k1s[j];
        xh2[1 + j]       = outm[OSC + j];
        xh2[1 + OSC + j] = k2s[j];
    }
}

// Six 2048-length dot products -> d_out[0..5] = [h1,k1,f1,h2,k2,f2].
__global__ void __launch_bounds__(256)
proj6_kernel(const float* __restrict__ ws,
             const float* __restrict__ hw, const float* __restrict__ hb,
             const float* __restrict__ kw, const float* __restrict__ kb,
             const float* __restrict__ fw, const float* __restrict__ fb,
             float* __restrict__ out) {
    __shared__ float red[256];
    const int o = blockIdx.x;
    const float* v; const float* w; const float* b;
    switch (o) {
        case 0: v = ws + WS_H1N; w = hw; b = hb; break;
        case 1: v = ws + WS_K1N; w = kw; b = kb; break;
        case 2: v = ws + WS_F1N; w = fw; b = fb; break;
        case 3: v = ws + WS_H2N; w = hw; b = hb; break;
        case 4: v = ws + WS_K2N; w = kw; b = kb; break;
        default: v = ws + WS_F2N; w = fw; b = fb; break;
    }
    float s = 0.0f;
    for (int j = threadIdx.x; j < OSC; j += 256) s += v[j] * w[j];
    red[threadIdx.x] = s;
    __syncthreads();
    for (int d = 128; d > 0; d >>= 1) {
        if (threadIdx.x < d) red[threadIdx.x] += red[threadIdx.x + d];
        __syncthreads();
    }
    if (threadIdx.x == 0) out[o] = red[0] + b[0];
}

extern "C" void kernel_launch(void* const* d_in, const int* in_sizes, int n_in,
                              void* d_out, int out_size, void* d_ws, size_t ws_size,
                              hipStream_t stream) {
    (void)in_sizes; (void)n_in; (void)out_size; (void)ws_size;
    const float* x      = (const float*)d_in[0];
    const float* m_s    = (const float*)d_in[1];
    const float* h1_s   = (const float*)d_in[2];
    const float* h2_s   = (const float*)d_in[3];
    const float* k1_s   = (const float*)d_in[4];
    const float* k2_s   = (const float*)d_in[5];
    const float* f1_s   = (const float*)d_in[6];
    const float* f2_s   = (const float*)d_in[7];
    const float* m_w_ih = (const float*)d_in[8];
    const float* m_w_hh = (const float*)d_in[9];
    const float* m_b_ih = (const float*)d_in[10];
    const float* m_b_hh = (const float*)d_in[11];
    const float* m_out_w = (const float*)d_in[12];
    const float* m_out_b = (const float*)d_in[13];
    const float* h_w_ih = (const float*)d_in[14];
    const float* h_w_hh = (const float*)d_in[15];
    const float* h_b_ih = (const float*)d_in[16];
    const float* h_b_hh = (const float*)d_in[17];
    const float* h_out_w = (const float*)d_in[18];
    const float* h_out_b = (const float*)d_in[19];
    const float* k_w_ih = (const float*)d_in[20];
    const float* k_w_hh = (const float*)d_in[21];
    const float* k_b_ih = (const float*)d_in[22];
    const float* k_b_hh = (const float*)d_in[23];
    const float* k_out_w = (const float*)d_in[24];
    const float* k_out_b = (const float*)d_in[25];
    const float* f_w_ih = (const float*)d_in[26];
    const float* f_w_hh = (const float*)d_in[27];
    const float* f_b_ih = (const float*)d_in[28];
    const float* f_b_hh = (const float*)d_in[29];
    const float* f_out_w = (const float*)d_in[30];
    const float* f_out_b = (const float*)d_in[31];

    float* ws = (float*)d_ws;

    prep1_kernel<<<32, 256, 0, stream>>>(x, h1_s, h2_s, f1_s, f2_s, ws);

    // master GRU: H=4096, Kin=4101
    gru16_kernel<<<MHID / 16, 192, 0, stream>>>(m_w_ih, m_w_hh, m_b_ih, m_b_hh,
                                                ws + WS_XM, m_s, ws + WS_MNEW,
                                                MHID, 5 + 2 * OSC);
    // knee GRUs (independent of m): H=2048, Kin=4097
    gru16_kernel<<<OSC / 16, 192, 0, stream>>>(k_w_ih, k_w_hh, k_b_ih, k_b_hh,
                                               ws + WS_XK1, k1_s, ws + WS_K1N,
                                               OSC, 1 + 2 * OSC);
    gru16_kernel<<<OSC / 16, 192, 0, stream>>>(k_w_ih, k_w_hh, k_b_ih, k_b_hh,
                                               ws + WS_XK2, k2_s, ws + WS_K2N,
                                               OSC, 1 + 2 * OSC);
    // foot GRUs: H=2048, Kin=2049
    gru16_kernel<<<OSC / 16, 192, 0, stream>>>(f_w_ih, f_w_hh, f_b_ih, f_b_hh,
                                               ws + WS_XF1, f1_s, ws + WS_F1N,
                                               OSC, 1 + OSC);
    gru16_kernel<<<OSC / 16, 192, 0, stream>>>(f_w_ih, f_w_hh, f_b_ih, f_b_hh,
                                               ws + WS_XF2, f2_s, ws + WS_F2N,
                                               OSC, 1 + OSC);

    // out_m = m_new @ m_out_w.T + m_out_b  (4096 rows, K=4096)
    gemv16_kernel<<<(2 * OSC) / 16, 256, 0, stream>>>(m_out_w, m_out_b,
                                                      ws + WS_MNEW, ws + WS_OUTM, MHID);

    prep2_kernel<<<32, 256, 0, stream>>>(x, k1_s, k2_s, ws);

    // hip GRUs: H=2048, Kin=4097
    gru16_kernel<<<OSC / 16, 192, 0, stream>>>(h_w_ih, h_w_hh, h_b_ih, h_b_hh,
                                               ws + WS_XH1, h1_s, ws + WS_H1N,
                                               OSC, 1 + 2 * OSC);
    gru16_kernel<<<OSC / 16, 192, 0, stream>>>(h_w_ih, h_w_hh, h_b_ih, h_b_hh,
                                               ws + WS_XH2, h2_s, ws + WS_H2N,
                                               OSC, 1 + 2 * OSC);

    proj6_kernel<<<6, 256, 0, stream>>>(ws, h_out_w, h_out_b, k_out_w, k_out_b,
                                        f_out_w, f_out_b, (float*)d_out);
}